// GraphSparseConvolution_52536039965271
// MI455X (gfx1250) — compile-verified
//
#include <hip/hip_runtime.h>
#include <hip/hip_bf16.h>

// GCN layer: out = relu( A_sp @ ( dropout(X_sp) @ W ) )
// N=100000, F=256, O=128, NNZ_X=NNZ_A=1.6M, keep_prob=0.9
//
// Strategy (MI455X, wave32, WMMA, LDS):
//  Phase 1: densify dropout(X) into Xd [N,F] fp32 in workspace (1 atomic / nnz)
//  Phase 2: h = Xd @ W via V_WMMA_F32_16X16X4_F32. W staged in LDS with a
//           K-pair-interleaved layout: element (k,n) at sW[(k>>1)*288 + 2n + (k&1)].
//           -> each B fragment is ONE conflict-free ds_load_b64 into an aligned
//              VGPR pair (no packing moves).
//  Phase 3: out += adj_val * h[col]  (wave-per-edge, coalesced 512B row loads,
//           f32 atomics -- unavoidable for random COO scatter)
//  Phase 4: relu in place
// Fallback (small ws): edge-parallel SpMM1 straight into h.

typedef __attribute__((ext_vector_type(2))) float v2f;
typedef __attribute__((ext_vector_type(8))) float v8f;

#define GCN_F 256
#define GCN_O 128
#define INV_KEEP (1.0f / 0.9f)
#define SW_GRP 288   // words per 2-K-row group: 288%64==32 -> lane halves use disjoint banks

__global__ __launch_bounds__(256) void zero_f32(float* __restrict__ p, long long n) {
  long long i = (long long)blockIdx.x * blockDim.x + threadIdx.x;
  long long stride = (long long)gridDim.x * blockDim.x;
  for (; i < n; i += stride) p[i] = 0.0f;
}

// Phase 1: scatter masked/scaled x_vals into dense Xd [N, F]
__global__ __launch_bounds__(256) void scatter_x(const float* __restrict__ xv,
                                                 const int* __restrict__ xr,
                                                 const int* __restrict__ xc,
                                                 const unsigned char* __restrict__ mask,
                                                 float* __restrict__ Xd, int nnz) {
  int i = blockIdx.x * blockDim.x + threadIdx.x;
  if (i >= nnz) return;
  if (mask[i]) {
    atomicAdd(&Xd[(size_t)xr[i] * GCN_F + xc[i]], xv[i] * INV_KEEP);
  }
}

// Phase 2: H[N,O] = Xd[N,F] @ W[F,O], fp32 WMMA 16x16x4, W staged in LDS.
// One wave per 16-row strip; wave holds all 8 (16-wide) col tiles => 64 acc VGPRs.
// A frag (16x4 f32): lane L: M = L&15, VGPR0/1 = K = 2*(L>>4) + {0,1}
// B frag (4x16 f32): lane L: N = L&15, VGPR0 = row K=2*(L>>4), VGPR1 = row K=2*(L>>4)+1
// C/D (16x16 f32):   lane L: N = L&15, VGPR r = row  r + 8*(L>>4)
__global__ __launch_bounds__(256) void gemm_xw_wmma(const float* __restrict__ Xd,
                                                    const float* __restrict__ W,
                                                    float* __restrict__ H,
                                                    int nRows, int nRowBlocks) {
  // 64 K-rows per chunk = 32 interleaved 2-row groups of SW_GRP words
  __shared__ float sW[32 * SW_GRP];          // 36864 B

  const int waveRaw   = (blockIdx.x * blockDim.x + threadIdx.x) >> 5;
  const bool waveLive = waveRaw < nRowBlocks;
  const int wave      = waveLive ? waveRaw : (nRowBlocks - 1);  // wave-uniform clamp
  const int lane  = threadIdx.x & 31;
  const int m     = lane & 15;
  const int khalf = lane >> 4;               // 0 or 1

  int rowA = wave * 16 + m;
  if (rowA >= nRows) rowA = nRows - 1;       // branchless clamp (stores are guarded)
  const float* arow = Xd + (size_t)rowA * GCN_F + 2 * khalf;

  v8f acc[8];
#pragma unroll
  for (int t = 0; t < 8; ++t) acc[t] = (v8f){0.f,0.f,0.f,0.f,0.f,0.f,0.f,0.f};

  for (int kc = 0; kc < GCN_F; kc += 64) {
    // Stage W[kc..kc+64) into LDS, interleaving K-row pairs:
    // (k,n) -> sW[(k>>1)*SW_GRP + 2n + (k&1)].  2048 float4 stores per chunk.
    for (int idx = threadIdx.x; idx < 32 * 64; idx += 256) {
      const int g = idx >> 6;                // 2-row group within chunk
      const int j = idx & 63;                // float4 index: covers n = 2j, 2j+1
      const float2 r0 = *(const float2*)(W + (size_t)(kc + 2 * g)     * GCN_O + 2 * j);
      const float2 r1 = *(const float2*)(W + (size_t)(kc + 2 * g + 1) * GCN_O + 2 * j);
      const float4 v  = {r0.x, r1.x, r0.y, r1.y};
      *(float4*)&sW[g * SW_GRP + 4 * j] = v; // byte off = g*1152 + 16j -> 16B aligned
    }
    __syncthreads();

    for (int k = 0; k < 64; k += 4) {
      v2f a = *(const v2f*)(arow + kc + k);  // A[rowA][kc + k + 2*khalf + {0,1}]
      const float* gbase = &sW[((k >> 1) + khalf) * SW_GRP];
#pragma unroll
      for (int t = 0; t < 8; ++t) {
        const int n = t * 16 + m;
        v2f b = *(const v2f*)(gbase + 2 * n);   // one conflict-free ds_load_b64
        acc[t] = __builtin_amdgcn_wmma_f32_16x16x4_f32(
            /*neg_a=*/false, a, /*neg_b=*/false, b,
            /*c_mod=*/(short)0, acc[t], /*reuse_a=*/false, /*reuse_b=*/false);
      }
    }
    __syncthreads();
  }

  if (waveLive) {
#pragma unroll
    for (int t = 0; t < 8; ++t) {
#pragma unroll
      for (int r = 0; r < 8; ++r) {
        const int grow = wave * 16 + r + 8 * khalf;
        if (grow < nRows) H[(size_t)grow * GCN_O + t * 16 + m] = acc[t][r];
      }
    }
  }
}

// Fallback SpMM1 (no dense ws): wave per nnz, lane covers 4 of 128 output cols.
__global__ __launch_bounds__(256) void spmm_x_direct(const float* __restrict__ xv,
                                                     const int* __restrict__ xr,
                                                     const int* __restrict__ xc,
                                                     const unsigned char* __restrict__ mask,
                                                     const float* __restrict__ W,
                                                     float* __restrict__ H, int nnz) {
  const int e = (blockIdx.x * blockDim.x + threadIdx.x) >> 5;
  if (e >= nnz) return;
  if (!mask[e]) return;                      // wave-uniform (scalar per edge)
  const int lane = threadIdx.x & 31;
  const float v  = xv[e] * INV_KEEP;
  const int row  = xr[e];
  const int col  = xc[e];
  const float4 wv = *(const float4*)(W + (size_t)col * GCN_O + lane * 4);
  float* h = H + (size_t)row * GCN_O + lane * 4;
  atomicAdd(h + 0, v * wv.x);
  atomicAdd(h + 1, v * wv.y);
  atomicAdd(h + 2, v * wv.z);
  atomicAdd(h + 3, v * wv.w);
}

// Phase 3: wave per adjacency edge: out[row] += val * h[col]
__global__ __launch_bounds__(256) void spmm_adj(const float* __restrict__ av,
                                                const int* __restrict__ ar,
                                                const int* __restrict__ ac,
                                                const float* __restrict__ H,
                                                float* __restrict__ out, int nnz) {
  const int e = (blockIdx.x * blockDim.x + threadIdx.x) >> 5;
  if (e >= nnz) return;
  const int lane = threadIdx.x & 31;
  const int row = ar[e];
  const int col = ac[e];
  const float v = av[e];
  const float4 hv = *(const float4*)(H + (size_t)col * GCN_O + lane * 4);
  float* o = out + (size_t)row * GCN_O + lane * 4;
  atomicAdd(o + 0, v * hv.x);
  atomicAdd(o + 1, v * hv.y);
  atomicAdd(o + 2, v * hv.z);
  atomicAdd(o + 3, v * hv.w);
}

__global__ __launch_bounds__(256) void relu_inplace(float* __restrict__ p, long long n) {
  long long i = (long long)blockIdx.x * blockDim.x + threadIdx.x;
  long long stride = (long long)gridDim.x * blockDim.x;
  for (; i < n; i += stride) {
    float x = p[i];
    p[i] = x > 0.0f ? x : 0.0f;
  }
}

extern "C" void kernel_launch(void* const* d_in, const int* in_sizes, int n_in,
                              void* d_out, int out_size, void* d_ws, size_t ws_size,
                              hipStream_t stream) {
  const float*         x_vals    = (const float*)d_in[0];
  const float*         W         = (const float*)d_in[1];   // [F, O]
  const float*         adj_vals  = (const float*)d_in[2];
  const int*           x_rows    = (const int*)d_in[3];
  const int*           x_cols    = (const int*)d_in[4];
  const int*           adj_rows  = (const int*)d_in[5];
  const int*           adj_cols  = (const int*)d_in[6];
  const unsigned char* keep_mask = (const unsigned char*)d_in[7]; // jax bool = 1 byte

  const int nnz_x = in_sizes[0];
  const int nnz_a = in_sizes[2];
  const int N     = out_size / GCN_O;         // 100000
  float* out = (float*)d_out;
  float* H   = (float*)d_ws;                  // [N, O]

  const long long nOut = (long long)N * GCN_O;
  const long long nXd  = (long long)N * GCN_F;
  const size_t needDense = (size_t)(nOut + nXd) * sizeof(float);
  const bool dense = ws_size >= needDense;

  const int T = 256;
  const int zgrid = 8192;

  // zero accumulation target for SpMM2 atomics (d_out is poisoned by harness)
  zero_f32<<<zgrid, T, 0, stream>>>(out, nOut);

  if (dense) {
    float* Xd = H + nOut;
    zero_f32<<<zgrid, T, 0, stream>>>(Xd, nXd);
    scatter_x<<<(nnz_x + T - 1) / T, T, 0, stream>>>(x_vals, x_rows, x_cols,
                                                     keep_mask, Xd, nnz_x);
    const int rowBlocks = (N + 15) / 16;      // 6250
    gemm_xw_wmma<<<(rowBlocks * 32 + T - 1) / T, T, 0, stream>>>(Xd, W, H, N, rowBlocks);
  } else {
    zero_f32<<<zgrid, T, 0, stream>>>(H, nOut);
    spmm_x_direct<<<((long long)nnz_x * 32 + T - 1) / T, T, 0, stream>>>(
        x_vals, x_rows, x_cols, keep_mask, W, H, nnz_x);
  }

  spmm_adj<<<((long long)nnz_a * 32 + T - 1) / T, T, 0, stream>>>(
      adj_vals, adj_rows, adj_cols, H, out, nnz_a);

  relu_inplace<<<zgrid, T, 0, stream>>>(out, nOut);
}